// Triplet_59588376264907
// MI455X (gfx1250) — compile-verified
//
#include <hip/hip_runtime.h>

typedef __attribute__((ext_vector_type(2))) float v2f;
typedef __attribute__((ext_vector_type(8))) float v8f;

#define N 384
#define D 256
#define MARGIN 1.0f

// ---------------------------------------------------------------------------
// Kernel 1: Gram tile via V_WMMA_F32_16X16X4_F32 + fused distance computation.
// One wave32 per 16x16 output tile. grid = (24, 24), block = 32.
//
// A-operand layout (16x4 f32, 2 VGPRs): lane L holds row (L&15),
//   K = 2*(L>>4) + {0,1}  -> a float2 from the row-major input row.
// B-operand (4x16, B[k][n] = A[j0+n][k]) has the identical per-lane gather
// pattern with the j-tile base. C/D: VGPR p, lane L -> M = p + 8*(L>>4),
//   Nc = L&15.
// ---------------------------------------------------------------------------
__global__ __launch_bounds__(32)
void triplet_gram_dist(const float* __restrict__ A, float* __restrict__ dmat) {
    const int bi   = blockIdx.x;            // i tile
    const int bj   = blockIdx.y;            // j tile
    const int lane = threadIdx.x;           // 0..31
    const int r    = lane & 15;
    const int half = lane >> 4;             // 0 or 1

    const float* __restrict__ rowI = A + (bi * 16 + r) * D;
    const float* __restrict__ rowJ = A + (bj * 16 + r) * D;

    // Per-lane row norm: lanes 0-15 -> i-rows, lanes 16-31 -> j-rows.
    const float* __restrict__ myrow = (half == 0) ? rowI : rowJ;
    float nrm = 0.0f;
    #pragma unroll 4
    for (int k = 0; k < D; k += 4) {
        float4 v = *reinterpret_cast<const float4*>(myrow + k);
        nrm = fmaf(v.x, v.x, nrm);
        nrm = fmaf(v.y, v.y, nrm);
        nrm = fmaf(v.z, v.z, nrm);
        nrm = fmaf(v.w, v.w, nrm);
    }

    // Gram tile: 64 steps of 16x16x4 fp32 WMMA over K = 256.
    v8f c = {};
    const int koff = 2 * half;
    #pragma unroll 8
    for (int k0 = 0; k0 < D; k0 += 4) {
        v2f a = *reinterpret_cast<const v2f*>(rowI + k0 + koff);
        v2f b = *reinterpret_cast<const v2f*>(rowJ + k0 + koff);
        c = __builtin_amdgcn_wmma_f32_16x16x4_f32(
                /*neg_a=*/false, a, /*neg_b=*/false, b,
                /*c_mod=*/(short)0, c, /*reuse_a=*/false, /*reuse_b=*/false);
    }

    // d[i,j] = max(norm_i + norm_j - 2*prod, 0)
    const float nj = __shfl(nrm, 16 + r, 32);   // norm of column row (j-tile)
    #pragma unroll
    for (int p = 0; p < 8; ++p) {
        const int   il = p + 8 * half;          // local i row of c[p]
        const float ni = __shfl(nrm, il, 32);   // lanes 0..15 hold i-norms
        float dist = fmaxf(ni + nj - 2.0f * c[p], 0.0f);
        dmat[(bi * 16 + il) * N + bj * 16 + r] = dist;
    }
}

// ---------------------------------------------------------------------------
// Kernel 2: per-anchor hinge reduction. One block per i (384 blocks, 256 thr).
// num_i = sum_{j: lab[j]!=lab[i]} sum_{k: lab[k]==lab[i]} max(d[i,k]+d[i,j]-1, 0)
// den_i = npos_i * (N - npos_i)
// Deterministic: fixed-order tree reduction, per-block partial outputs.
// ---------------------------------------------------------------------------
__global__ __launch_bounds__(256)
void triplet_hinge(const float* __restrict__ dmat, const int* __restrict__ labels,
                   float* __restrict__ num_part, float* __restrict__ den_part) {
    __shared__ float s_d[N];
    __shared__ int   s_lab[N];
    __shared__ float s_pd[N];     // d values of positive set (compacted)
    __shared__ int   s_npos;
    __shared__ float s_red[256];

    const int i = blockIdx.x;
    const int t = threadIdx.x;

    for (int k = t; k < N; k += 256) {
        s_d[k]   = dmat[i * N + k];
        s_lab[k] = labels[k];
    }
    __syncthreads();

    const int lab_i = s_lab[i];
    if (t == 0) {
        int cnt = 0;
        for (int k = 0; k < N; ++k)
            if (s_lab[k] == lab_i) s_pd[cnt++] = s_d[k];
        s_npos = cnt;
    }
    __syncthreads();
    const int npos = s_npos;

    float local = 0.0f;
    for (int j = t; j < N; j += 256) {
        if (s_lab[j] != lab_i) {
            const float base = s_d[j] - MARGIN;
            float acc = 0.0f;
            for (int p = 0; p < npos; ++p)
                acc += fmaxf(s_pd[p] + base, 0.0f);
            local += acc;
        }
    }
    s_red[t] = local;
    __syncthreads();
    #pragma unroll
    for (int s = 128; s > 0; s >>= 1) {
        if (t < s) s_red[t] += s_red[t + s];
        __syncthreads();
    }
    if (t == 0) {
        num_part[i] = s_red[0];
        den_part[i] = (float)(npos * (N - npos));
    }
}

// ---------------------------------------------------------------------------
// Kernel 3: fold 384 partials -> out[0] = num / den. One block, 256 threads.
// ---------------------------------------------------------------------------
__global__ __launch_bounds__(256)
void triplet_finalize(const float* __restrict__ num_part,
                      const float* __restrict__ den_part,
                      float* __restrict__ out) {
    __shared__ float sn[256];
    __shared__ float sd[256];
    const int t = threadIdx.x;
    float n = num_part[t];
    float d = den_part[t];
    if (t < N - 256) { n += num_part[t + 256]; d += den_part[t + 256]; }
    sn[t] = n;
    sd[t] = d;
    __syncthreads();
    #pragma unroll
    for (int s = 128; s > 0; s >>= 1) {
        if (t < s) { sn[t] += sn[t + s]; sd[t] += sd[t + s]; }
        __syncthreads();
    }
    if (t == 0) out[0] = sn[0] / sd[0];
}

extern "C" void kernel_launch(void* const* d_in, const int* in_sizes, int n_in,
                              void* d_out, int out_size, void* d_ws, size_t ws_size,
                              hipStream_t stream) {
    const float* input  = (const float*)d_in[0];   // [384, 256] f32
    const int*   labels = (const int*)d_in[1];     // [384] i32
    float*       out    = (float*)d_out;           // scalar f32

    float* dmat     = (float*)d_ws;                // 384*384 floats
    float* num_part = dmat + N * N;                // 384 floats
    float* den_part = num_part + N;                // 384 floats

    dim3 ggrid(N / 16, N / 16);
    triplet_gram_dist<<<ggrid, 32, 0, stream>>>(input, dmat);
    triplet_hinge<<<N, 256, 0, stream>>>(dmat, labels, num_part, den_part);
    triplet_finalize<<<1, 256, 0, stream>>>(num_part, den_part, out);
}